// BlockGivensRotation_90675349553344
// MI455X (gfx1250) — compile-verified
//
#include <hip/hip_runtime.h>

typedef __attribute__((ext_vector_type(2))) float v2f;
typedef __attribute__((ext_vector_type(8))) float v8f;

#define OUTF 8192
#define INF  8192
#define BLK  128
#define NBLK 64
#define NSWP 8

// ---------------------------------------------------------------------------
// Kernel 1: build per-block rotation matrices R_b (128x128) from angles.
// One workgroup per block, one thread per row of R (rows are independent:
// the Givens carry recurrence only mixes columns within a row).
// Row kept in registers (inner 127-step loop fully unrolled -> const indices).
// Output written to d_ws in WMMA-B swizzled layout:
//   ws[b*16384 + (k>>2)*512 + n*4 + (k&3)] = R_b[k][n]
// so kernel 2 lane l reads its B fragment as one contiguous 8-byte ds load.
// ---------------------------------------------------------------------------
__global__ __launch_bounds__(128) void givens_build_r(
    const float* __restrict__ angles, float* __restrict__ Rws) {
  __shared__ float cs[NSWP * (BLK - 1)];
  __shared__ float sn[NSWP * (BLK - 1)];
  const int b = blockIdx.x;
  const int t = threadIdx.x;

  // Cooperative trig precompute: angles is [NB][S][B-1] row-major.
  for (int i = t; i < NSWP * (BLK - 1); i += 128) {
    float a = angles[b * (NSWP * (BLK - 1)) + i];
    cs[i] = __builtin_cosf(a);
    sn[i] = __builtin_sinf(a);
  }
  __syncthreads();

  // Row t of R starts as e_t.
  float r[BLK];
#pragma unroll
  for (int n = 0; n < BLK; ++n) r[n] = (n == t) ? 1.0f : 0.0f;

  for (int s = 0; s < NSWP; ++s) {
    const float* c = cs + s * (BLK - 1);
    const float* si = sn + s * (BLK - 1);
    float carry = r[0];
#pragma unroll
    for (int i = 0; i < BLK - 1; ++i) {
      float xj = r[i + 1];
      float ci = c[i];
      float sii = si[i];
      r[i] = ci * carry - sii * xj;      // final column i
      carry = sii * carry + ci * xj;     // becomes current column i+1
    }
    r[BLK - 1] = carry;
  }

  // Write swizzled B layout: (k4, n, k&3) with k = t.
  float* dst = Rws + (size_t)b * (BLK * BLK) + (t >> 2) * 512 + (t & 3);
#pragma unroll
  for (int n = 0; n < BLK; ++n) dst[n * 4] = r[n];
}

// ---------------------------------------------------------------------------
// Kernel 2: out[:, b*128 : b*128+128] = W_block @ R_b via V_WMMA_F32_16X16X4_F32.
// Grid: (64 blocks, 64 row-tiles). 256 threads = 8 waves; wave w owns rows
// [tile*128 + w*16, +16) and all 8 N-subtiles (8 fp32 accumulators).
// R_b staged in LDS (64 KB, swizzled); A fragments stream from global
// (b64 per lane; each 128B line feeds 8 consecutive K-steps through L0).
// ---------------------------------------------------------------------------
__global__ __launch_bounds__(256) void givens_gemm(
    const float* __restrict__ w, const float* __restrict__ Rws,
    float* __restrict__ out) {
  __shared__ __align__(16) float Rs[BLK * BLK];  // swizzled [k4][n][k&3]

  const int b = blockIdx.x;        // which 128-column block
  const int mrow0 = blockIdx.y * 128;
  const int tid = threadIdx.x;

  // Copy this block's swizzled R (16384 floats) into LDS with float4s.
  {
    const float4* src = (const float4*)(Rws + (size_t)b * (BLK * BLK));
    float4* dst = (float4*)Rs;
#pragma unroll
    for (int i = 0; i < 16; ++i) dst[i * 256 + tid] = src[i * 256 + tid];
  }
  __syncthreads();

  const int wave = tid >> 5;
  const int lane = tid & 31;
  const int half = lane >> 4;   // 0: K0/K1 half, 1: K2/K3 half
  const int lrow = lane & 15;   // M (for A) / N (for B,C) index within tile

  // A fragment source row (lanes 0-15 and 16-31 both cover M=0..15).
  const size_t rowbase = (size_t)(mrow0 + wave * 16 + lrow) * INF + b * BLK;

  v8f acc[8];
#pragma unroll
  for (int j = 0; j < 8; ++j) acc[j] = v8f{};

  for (int k4 = 0; k4 < 32; ++k4) {
    const int k = k4 * 4;
    // A: W[m][k + 2*half .. +1]  (8B, 8-byte aligned)
    const v2f av = *(const v2f*)(w + rowbase + k + 2 * half);
    // B fragments: swizzled LDS, contiguous 8B per lane, conflict-free banks.
    const float* rb = Rs + k4 * 512 + lrow * 4 + 2 * half;
#pragma unroll
    for (int j = 0; j < 8; ++j) {
      const v2f bv = *(const v2f*)(rb + j * 64);
      acc[j] = __builtin_amdgcn_wmma_f32_16x16x4_f32(
          false, av, false, bv, (short)0, acc[j], false, false);
    }
  }

  // Store: C/D layout — VGPR v, lane l: row M = v + 8*(l>=16), col N = l%16.
  const int c0 = b * BLK;
#pragma unroll
  for (int v = 0; v < 8; ++v) {
    float* op = out + (size_t)(mrow0 + wave * 16 + v + 8 * half) * INF + c0 + lrow;
#pragma unroll
    for (int j = 0; j < 8; ++j) op[j * 16] = acc[j][v];
  }
}

extern "C" void kernel_launch(void* const* d_in, const int* in_sizes, int n_in,
                              void* d_out, int out_size, void* d_ws, size_t ws_size,
                              hipStream_t stream) {
  (void)in_sizes; (void)n_in; (void)out_size; (void)ws_size;
  const float* w      = (const float*)d_in[0];
  const float* angles = (const float*)d_in[1];
  float* out = (float*)d_out;
  float* Rws = (float*)d_ws;  // 64 * 128 * 128 * 4 = 4 MB scratch

  givens_build_r<<<dim3(NBLK), dim3(128), 0, stream>>>(angles, Rws);
  givens_gemm<<<dim3(NBLK, OUTF / 128), dim3(256), 0, stream>>>(w, Rws, out);
}